// Embedding_62371515072547
// MI455X (gfx1250) — compile-verified
//
#include <hip/hip_runtime.h>
#include <cstdint>

// Embedding gather: out[r, :] = W[x[r], :], r in [0, 4096), EMBED_DIM = 768 fp32.
// Memory-bound: ~12.6 MB read + 12.6 MB write => ~1.1 us at 23.3 TB/s HBM.
// Implemented on the CDNA5 async global<->LDS data mover (ASYNCcnt-tracked).
//
// Mapping: one wave32 per row. Row = 3072 B = 192 float4 chunks = 32 lanes x 6
// chunks. The async ops apply INST_OFFSET to BOTH the global and LDS address
// (CDNA5 ISA), so each lane needs exactly one 64-bit global
// address and one 32-bit LDS offset; the 6 chunks are issued as an s_clause'd
// burst with offset:0,512,...,2560 immediates.

#define ROW_BYTES        3072                     // 768 * 4
#define ROWS_PER_BLOCK   8                        // one row per wave32
#define THREADS          256                      // 8 wave32s

__global__ __launch_bounds__(THREADS)
void embed_gather_async(const int* __restrict__ x,
                        const float* __restrict__ W,
                        float* __restrict__ out,
                        int n_rows)
{
    __shared__ __align__(16) unsigned char tile[ROWS_PER_BLOCK * ROW_BYTES];

    // Low 32 bits of a generic pointer into LDS == group-segment byte offset,
    // which is what the async ops' LDS-address VGPR expects.
    const uint32_t lds_base = (uint32_t)(uintptr_t)(void*)&tile[0];

    const int lane = (int)(threadIdx.x & 31u);
    const int wv   = (int)(threadIdx.x >> 5);     // wave id == row within block
    const int row  = blockIdx.x * ROWS_PER_BLOCK + wv;

    const uint32_t lds = lds_base + (uint32_t)wv * ROW_BYTES + (uint32_t)lane * 16u;

    // ---- Phase 1: async gather, global -> LDS (6 x 16 B per lane) ----
    if (row < n_rows) {
        const int      idx = x[row];
        const uint64_t src = (uint64_t)(uintptr_t)W
                           + (uint64_t)idx * ROW_BYTES
                           + (uint64_t)lane * 16u;
        asm volatile(
            "s_clause 0x5\n\t"
            "global_load_async_to_lds_b128 %0, %1, off\n\t"
            "global_load_async_to_lds_b128 %0, %1, off offset:512\n\t"
            "global_load_async_to_lds_b128 %0, %1, off offset:1024\n\t"
            "global_load_async_to_lds_b128 %0, %1, off offset:1536\n\t"
            "global_load_async_to_lds_b128 %0, %1, off offset:2048\n\t"
            "global_load_async_to_lds_b128 %0, %1, off offset:2560"
            :: "v"(lds), "v"(src)
            : "memory");
    }

    // Each lane reads back exactly the LDS bytes it wrote (same wave), so no
    // barrier is needed — only this wave's ASYNCcnt must drain.
    asm volatile("s_wait_asynccnt 0" ::: "memory");

    // ---- Phase 2: async write-out, LDS -> global (fully contiguous) ----
    if (row < n_rows) {
        const uint64_t dst = (uint64_t)(uintptr_t)out
                           + (uint64_t)row * ROW_BYTES
                           + (uint64_t)lane * 16u;
        asm volatile(
            "s_clause 0x5\n\t"
            "global_store_async_from_lds_b128 %0, %1, off\n\t"
            "global_store_async_from_lds_b128 %0, %1, off offset:512\n\t"
            "global_store_async_from_lds_b128 %0, %1, off offset:1024\n\t"
            "global_store_async_from_lds_b128 %0, %1, off offset:1536\n\t"
            "global_store_async_from_lds_b128 %0, %1, off offset:2048\n\t"
            "global_store_async_from_lds_b128 %0, %1, off offset:2560"
            :: "v"(dst), "v"(lds)
            : "memory");
    }

    // Make async-store completion explicit (s_endpgm also implies wait-idle).
    asm volatile("s_wait_asynccnt 0" ::: "memory");
}

extern "C" void kernel_launch(void* const* d_in, const int* in_sizes, int n_in,
                              void* d_out, int out_size, void* d_ws, size_t ws_size,
                              hipStream_t stream)
{
    const int*   x   = (const int*)d_in[0];    // [BATCH*POS] int32 token ids
    const float* W   = (const float*)d_in[1];  // [32000, 768] fp32 table
    float*       out = (float*)d_out;          // [BATCH*POS, 768] fp32

    const int n_rows = in_sizes[0];            // 4096
    const int blocks = (n_rows + ROWS_PER_BLOCK - 1) / ROWS_PER_BLOCK;

    embed_gather_async<<<blocks, THREADS, 0, stream>>>(x, W, out, n_rows);
}